// AttentionDecoder_82532091560494
// MI455X (gfx1250) — compile-verified
//
#include <hip/hip_runtime.h>
#include <hip/hip_bf16.h>
#include <math.h>

#define HIDDEN 1024
#define VOCAB  50257
#define MAXLEN 512

typedef __attribute__((ext_vector_type(2))) float v2f;
typedef __attribute__((ext_vector_type(8))) float v8f;

// ---------------------------------------------------------------------------
// K0: gather embedding row, build cat = [embed, h0] and cat2 lower half.
// ---------------------------------------------------------------------------
__global__ void prep_kernel(const int* __restrict__ token,
                            const float* __restrict__ emb,
                            const float* __restrict__ h0,
                            float* __restrict__ cat,    // 2048
                            float* __restrict__ cat2)   // 2048 (lower half here)
{
    int i = blockIdx.x * blockDim.x + threadIdx.x;
    if (i >= HIDDEN) return;
    float e = emb[(size_t)token[0] * HIDDEN + i];
    cat[i]          = e;
    cat2[i]         = e;
    cat[HIDDEN + i] = h0[i];
}

// ---------------------------------------------------------------------------
// Wave-per-16-rows fp32 GEMV on the WMMA pipe: y[r] = dot(W[r,:], x) + b[r].
//
// A = 16x4 fp32 tile of W (ISA layout: lanes 0-15 hold K=0,1; lanes 16-31
// hold K=2,3), B = x[k..k+3] broadcast across all 16 columns, so every
// column of the 16x16 f32 D tile holds the same 16 dot products; lanes 0
// and 16 extract column 0.
//
// K is a compile-time constant (unroll-by-4/clausing); x is staged in LDS
// (DS-pipe broadcast) so VMEM is weight-streaming only; two independent
// accumulators break the WMMA D->C dependency chain; NT = non-temporal
// weight loads for the read-once 206 MB out_W stream (preserves L2 for the
// small weights across graph replays).
// ---------------------------------------------------------------------------
template <int K, bool NT, int ACT>
__global__ void gemv_wmma_kernel(const float* __restrict__ W,
                                 const float* __restrict__ x,
                                 const float* __restrict__ bias,
                                 float* __restrict__ y,
                                 int rows)
{
    __shared__ float xs[K];
    for (int i = threadIdx.x; i < K; i += blockDim.x) xs[i] = x[i];
    __syncthreads();

    const int wave   = (blockIdx.x * blockDim.x + threadIdx.x) >> 5;
    const int lane   = threadIdx.x & 31;
    const int ntiles = (rows + 15) >> 4;
    if (wave >= ntiles) return;          // wave-uniform: EXEC stays all-1s below

    const int row0  = wave << 4;
    const int mrow  = lane & 15;         // A-matrix row this lane feeds
    const int khalf = (lane >> 4) << 1;  // lanes 0-15: K=0,1 ; lanes 16-31: K=2,3

    int arow = row0 + mrow;
    if (arow > rows - 1) arow = rows - 1;   // clamp loads in the ragged last tile
    const float* __restrict__ wrow = W + (size_t)arow * K + khalf;
    const float* __restrict__ xk   = xs + khalf;

    v8f c0 = {0.f, 0.f, 0.f, 0.f, 0.f, 0.f, 0.f, 0.f};
    v8f c1 = {0.f, 0.f, 0.f, 0.f, 0.f, 0.f, 0.f, 0.f};

    #pragma unroll 4
    for (int k = 0; k < K; k += 8) {
        v2f a0, a1, b0, b1;
        if constexpr (NT) {
            a0 = __builtin_nontemporal_load(reinterpret_cast<const v2f*>(wrow + k));
            a1 = __builtin_nontemporal_load(reinterpret_cast<const v2f*>(wrow + k + 4));
        } else {
            a0 = *reinterpret_cast<const v2f*>(wrow + k);
            a1 = *reinterpret_cast<const v2f*>(wrow + k + 4);
        }
        b0 = *reinterpret_cast<const v2f*>(xk + k);
        b1 = *reinterpret_cast<const v2f*>(xk + k + 4);
        c0 = __builtin_amdgcn_wmma_f32_16x16x4_f32(
                 false, a0, false, b0, (short)0, c0, false, false);
        c1 = __builtin_amdgcn_wmma_f32_16x16x4_f32(
                 false, a1, false, b1, (short)0, c1, false, false);
    }
    v8f c = c0 + c1;

    // D layout: VGPR r -> row r (lanes 0-15) / row r+8 (lanes 16-31), N = lane%16.
    if (mrow == 0) {
        const int mbase = (lane >> 4) << 3;   // 0 or 8
        #pragma unroll
        for (int i = 0; i < 8; ++i) {
            int r = row0 + mbase + i;
            if (r < rows) {
                float v = c[i] + bias[r];
                if (ACT == 1) v = fmaxf(v, 0.f);
                y[r] = v;
            }
        }
    }
}

// ---------------------------------------------------------------------------
// K2: softmax over the 512 attention logits (single block of 512 threads).
// ---------------------------------------------------------------------------
__global__ void softmax512_kernel(const float* __restrict__ logits,
                                  float* __restrict__ w_ws,
                                  float* __restrict__ w_out)
{
    __shared__ float red[MAXLEN];
    int t = threadIdx.x;
    float v = logits[t];
    red[t] = v;
    __syncthreads();
    for (int s = MAXLEN / 2; s > 0; s >>= 1) {
        if (t < s) red[t] = fmaxf(red[t], red[t + s]);
        __syncthreads();
    }
    float m = red[0];
    __syncthreads();
    float e = expf(v - m);
    red[t] = e;
    __syncthreads();
    for (int s = MAXLEN / 2; s > 0; s >>= 1) {
        if (t < s) red[t] += red[t + s];
        __syncthreads();
    }
    float w = e / red[0];
    w_ws[t]  = w;
    w_out[t] = w;   // third output of the reference
}

// ---------------------------------------------------------------------------
// K3: attn_applied[j] = sum_l w[l] * enc[l][j]  (coalesced over j).
// ---------------------------------------------------------------------------
__global__ void attn_apply_kernel(const float* __restrict__ w,
                                  const float* __restrict__ enc,
                                  float* __restrict__ cat2_hi)
{
    __shared__ float ws[MAXLEN];
    for (int i = threadIdx.x; i < MAXLEN; i += blockDim.x) ws[i] = w[i];
    __syncthreads();
    int j = blockIdx.x * blockDim.x + threadIdx.x;
    if (j >= HIDDEN) return;
    float acc = 0.f;
    for (int l = 0; l < MAXLEN; ++l) acc += ws[l] * enc[(size_t)l * HIDDEN + j];
    cat2_hi[j] = acc;
}

// ---------------------------------------------------------------------------
// K6: fuse GRU gates -> h_new (to workspace and to d_out slice).
// ---------------------------------------------------------------------------
__global__ void gru_combine_kernel(const float* __restrict__ gi,
                                   const float* __restrict__ gh,
                                   const float* __restrict__ h0,
                                   float* __restrict__ hnew_ws,
                                   float* __restrict__ hnew_out)
{
    int i = blockIdx.x * blockDim.x + threadIdx.x;
    if (i >= HIDDEN) return;
    float r = 1.f / (1.f + expf(-(gi[i] + gh[i])));
    float z = 1.f / (1.f + expf(-(gi[HIDDEN + i] + gh[HIDDEN + i])));
    float n = tanhf(gi[2 * HIDDEN + i] + r * gh[2 * HIDDEN + i]);
    float h = (1.f - z) * n + z * h0[i];
    hnew_ws[i]  = h;
    hnew_out[i] = h;
}

// ---------------------------------------------------------------------------
// K8: max and log-sum-exp over the 50257 logits (single block, 1024 threads).
// ---------------------------------------------------------------------------
__global__ void lse_kernel(const float* __restrict__ logits,
                           float* __restrict__ red_out /* [max, logZ] */)
{
    __shared__ float red[1024];
    int t = threadIdx.x;
    float m = -INFINITY;
    for (int v = t; v < VOCAB; v += 1024) m = fmaxf(m, logits[v]);
    red[t] = m;
    __syncthreads();
    for (int s = 512; s > 0; s >>= 1) {
        if (t < s) red[t] = fmaxf(red[t], red[t + s]);
        __syncthreads();
    }
    m = red[0];
    __syncthreads();
    float sum = 0.f;
    for (int v = t; v < VOCAB; v += 1024) sum += expf(logits[v] - m);
    red[t] = sum;
    __syncthreads();
    for (int s = 512; s > 0; s >>= 1) {
        if (t < s) red[t] += red[t + s];
        __syncthreads();
    }
    if (t == 0) { red_out[0] = m; red_out[1] = logf(red[0]); }
}

// ---------------------------------------------------------------------------
// K9: in-place log-softmax: logp[v] = logits[v] - max - logZ.
// ---------------------------------------------------------------------------
__global__ void logp_kernel(float* __restrict__ logits,
                            const float* __restrict__ red)
{
    int v = blockIdx.x * blockDim.x + threadIdx.x;
    if (v >= VOCAB) return;
    logits[v] = logits[v] - red[0] - red[1];
}

// ---------------------------------------------------------------------------
extern "C" void kernel_launch(void* const* d_in, const int* in_sizes, int n_in,
                              void* d_out, int out_size, void* d_ws, size_t ws_size,
                              hipStream_t stream)
{
    (void)in_sizes; (void)n_in; (void)out_size; (void)ws_size;

    const int*   token  = (const int*)  d_in[0];
    const float* h0     = (const float*)d_in[1];
    const float* enc    = (const float*)d_in[2];
    const float* emb    = (const float*)d_in[3];
    const float* attn_W = (const float*)d_in[4];
    const float* attn_b = (const float*)d_in[5];
    const float* comb_W = (const float*)d_in[6];
    const float* comb_b = (const float*)d_in[7];
    const float* W_ih   = (const float*)d_in[8];
    const float* W_hh   = (const float*)d_in[9];
    const float* b_ih   = (const float*)d_in[10];
    const float* b_hh   = (const float*)d_in[11];
    const float* out_W  = (const float*)d_in[12];
    const float* out_b  = (const float*)d_in[13];

    float* out = (float*)d_out;
    float* out_logp = out;                        // 50257
    float* out_h    = out + VOCAB;                // 1024
    float* out_attn = out + VOCAB + HIDDEN;       // 512

    // workspace layout (floats), total ~53 KB
    float* f        = (float*)d_ws;
    float* cat      = f;               // 2048
    float* cat2     = f + 2048;        // 2048
    float* attn_lg  = f + 4096;        // 512
    float* attn_w   = f + 4608;        // 512
    float* xbuf     = f + 5120;        // 1024
    float* gi       = f + 6144;        // 3072
    float* gh       = f + 9216;        // 3072
    float* hnew     = f + 12288;       // 1024
    float* red      = f + 13312;       // 2

    // 1) embedding gather + concat
    prep_kernel<<<(HIDDEN + 255) / 256, 256, 0, stream>>>(token, emb, h0, cat, cat2);

    // 2) attention logits: (512 x 2048) GEMV -> 32 waves, 8 waves/block
    gemv_wmma_kernel<2 * HIDDEN, false, 0><<<4, 256, 0, stream>>>(
        attn_W, cat, attn_b, attn_lg, MAXLEN);

    // 3) softmax over 512
    softmax512_kernel<<<1, MAXLEN, 0, stream>>>(attn_lg, attn_w, out_attn);

    // 4) attn_applied = w @ encoder_outputs  -> cat2 upper half
    attn_apply_kernel<<<(HIDDEN + 255) / 256, 256, 0, stream>>>(attn_w, enc, cat2 + HIDDEN);

    // 5) combine + ReLU: (1024 x 2048) GEMV -> 64 waves
    gemv_wmma_kernel<2 * HIDDEN, false, 1><<<8, 256, 0, stream>>>(
        comb_W, cat2, comb_b, xbuf, HIDDEN);

    // 6) GRU gate GEMVs: (3072 x 1024) each -> 192 waves each
    gemv_wmma_kernel<HIDDEN, false, 0><<<24, 256, 0, stream>>>(
        W_ih, xbuf, b_ih, gi, 3 * HIDDEN);
    gemv_wmma_kernel<HIDDEN, false, 0><<<24, 256, 0, stream>>>(
        W_hh, h0, b_hh, gh, 3 * HIDDEN);

    // 7) gate fusion -> h_new
    gru_combine_kernel<<<(HIDDEN + 255) / 256, 256, 0, stream>>>(gi, gh, h0, hnew, out_h);

    // 8) output projection: (50257 x 1024) GEMV, non-temporal weight stream,
    //    logits straight into d_out. 3142 waves -> 393 blocks of 8 waves.
    {
        int ntiles = (VOCAB + 15) / 16;
        int blocks = (ntiles + 7) / 8;
        gemv_wmma_kernel<HIDDEN, true, 0><<<blocks, 256, 0, stream>>>(
            out_W, hnew, out_b, out_logp, VOCAB);
    }

    // 9) log-softmax statistics, then in-place logp
    lse_kernel<<<1, 1024, 0, stream>>>(out_logp, red);
    logp_kernel<<<(VOCAB + 255) / 256, 256, 0, stream>>>(out_logp, red);
}